// GraphSAGE_30571577213476
// MI455X (gfx1250) — compile-verified
//
#include <hip/hip_runtime.h>
#include <hip/hip_bf16.h>

typedef float v2f __attribute__((ext_vector_type(2)));
typedef float v8f __attribute__((ext_vector_type(8)));

#define LDSB_STRIDE_E 132   // 128 K + 4 pad -> conflict-free b64 LDS loads
#define LDSB_STRIDE_N 260   // 256 K + 4 pad
#define TPW_E 4             // edge tiles per wave
#define TPW_N 2             // node tiles per wave

extern __shared__ float ldsb[];

// ---------------------------------------------------------------------------
// Edge pooling: pooled[e,:] = relu((x[src[e]] * (1+coef*wt[e])) @ W^T + b)
// then agg[dst[e],:] = max(agg, pooled)  via integer atomicMax on float bits.
// One wave = 16 edges x 128 outputs per tile, TPW_E tiles per wave.
// W staged in LDS; k-loop software-pipelined (B frags + A row one step ahead).
// ---------------------------------------------------------------------------
__global__ __launch_bounds__(256) void edge_pool_kernel(
    const float* __restrict__ x,     // [N,128]
    const int*   __restrict__ src,   // [E]
    const int*   __restrict__ dst,   // [E]
    const float* __restrict__ wt,    // [E]
    const float* __restrict__ pw,    // [128,128] row-major (out,in)
    const float* __restrict__ pb,    // [128]
    const float* __restrict__ coefp, // [1]
    float*       __restrict__ agg,   // [N,128], zero-initialized
    int ntiles)
{
    const int tid = threadIdx.x;
    // stage W into LDS as [n][k] with padded stride
    for (int i = tid; i < 128 * 128; i += 256) {
        int n = i >> 7, k = i & 127;
        ldsb[n * LDSB_STRIDE_E + k] = pw[i];
    }
    __syncthreads();

    const int wave = tid >> 5;
    const int lane = tid & 31;
    const int m    = lane & 15;   // row within tile (edges) / B column
    const int hi   = lane >> 4;   // half-wave: K offset +2
    const float coef = coefp[0];

    float bias[8];
#pragma unroll
    for (int nt = 0; nt < 8; ++nt) bias[nt] = pb[nt * 16 + m];

    const int tbase = (blockIdx.x * 8 + wave) * TPW_E;

#pragma unroll 1
    for (int it = 0; it < TPW_E; ++it) {
        const int tile = tbase + it;
        if (tile >= ntiles) break;
        const int e0 = tile * 16;

        const int   srcm  = src[e0 + m];
        const int   dstm  = dst[e0 + m];
        const float scale = 1.0f + coef * wt[e0 + m];
        const float* arow = x + (long)srcm * 128 + 2 * hi;

        // prefetch next tile's gathered rows (512B each -> 4 x 128B lines)
        if (it + 1 < TPW_E && tile + 1 < ntiles) {
            const int srcn = src[(tile + 1) * 16 + m];
            const float* prow = x + (long)srcn * 128;
            __builtin_prefetch(prow,      0, 3);
            __builtin_prefetch(prow + 32, 0, 3);
            __builtin_prefetch(prow + 64, 0, 3);
            __builtin_prefetch(prow + 96, 0, 3);
        }

        const v8f zacc = {0.f, 0.f, 0.f, 0.f, 0.f, 0.f, 0.f, 0.f};
        v8f acc[8];
#pragma unroll
        for (int nt = 0; nt < 8; ++nt) acc[nt] = zacc;

        // --- software-pipelined k-loop: loads run one step ahead of WMMAs ---
        v2f bf[2][8];
#pragma unroll
        for (int nt = 0; nt < 8; ++nt)
            bf[0][nt] = *(const v2f*)(&ldsb[(nt * 16 + m) * LDSB_STRIDE_E + 2 * hi]);
        v2f a = *(const v2f*)(arow);
        a *= scale;

#pragma unroll
        for (int t = 0; t < 32; ++t) {
            const int cur = t & 1, nxt = cur ^ 1;
            v2f an = a;
            if (t < 31) {
                const int kk = 4 * (t + 1) + 2 * hi;
#pragma unroll
                for (int nt = 0; nt < 8; ++nt)
                    bf[nxt][nt] = *(const v2f*)(&ldsb[(nt * 16 + m) * LDSB_STRIDE_E + kk]);
                an = *(const v2f*)(arow + 4 * (t + 1));
                an *= scale;
            }
            __builtin_amdgcn_sched_barrier(0);
#pragma unroll
            for (int nt = 0; nt < 8; ++nt)
                acc[nt] = __builtin_amdgcn_wmma_f32_16x16x4_f32(
                    false, a, false, bf[cur][nt], (short)0, acc[nt], false, false);
            a = an;
        }

        // D layout: VGPR r, lanes 0-15 -> M=r, lanes 16-31 -> M=r+8; N = nt*16+m
#pragma unroll
        for (int r = 0; r < 8; ++r) {
            const int dr = __shfl(dstm, r + 8 * hi, 32);
            float* aggrow = agg + (long)dr * 128 + m;
#pragma unroll
            for (int nt = 0; nt < 8; ++nt) {
                float v = acc[nt][r] + bias[nt];
                v = fmaxf(v, 0.0f);
                // exact float max for v >= 0 with 0-initialized buffer
                atomicMax((int*)(aggrow + nt * 16), __float_as_int(v));
            }
        }
    }
}

// ---------------------------------------------------------------------------
// Node update: h[v,:] = relu(concat(x[v], agg[v]) @ Wf^T + bf),  Wf [128,256]
// One wave = 16 nodes x 128 outputs per tile, TPW_N tiles per wave, K = 256.
// ---------------------------------------------------------------------------
__global__ __launch_bounds__(256) void node_update_kernel(
    const float* __restrict__ x,    // [N,128]
    const float* __restrict__ agg,  // [N,128]
    const float* __restrict__ fw,   // [128,256] row-major (out,in)
    const float* __restrict__ fb,   // [128]
    float*       __restrict__ h,    // [N,128]
    int ntiles)
{
    const int tid = threadIdx.x;
    for (int i = tid; i < 128 * 256; i += 256) {
        int n = i >> 8, k = i & 255;
        ldsb[n * LDSB_STRIDE_N + k] = fw[i];
    }
    __syncthreads();

    const int wave = tid >> 5;
    const int lane = tid & 31;
    const int m    = lane & 15;
    const int hi   = lane >> 4;

    float bias[8];
#pragma unroll
    for (int nt = 0; nt < 8; ++nt) bias[nt] = fb[nt * 16 + m];

    const int tbase = (blockIdx.x * 8 + wave) * TPW_N;

#pragma unroll 1
    for (int it = 0; it < TPW_N; ++it) {
        const int tile = tbase + it;
        if (tile >= ntiles) break;
        const int v0 = tile * 16;

        const float* px = x   + (long)(v0 + m) * 128 + 2 * hi;
        const float* pa = agg + (long)(v0 + m) * 128 + 2 * hi;

        const v8f zacc = {0.f, 0.f, 0.f, 0.f, 0.f, 0.f, 0.f, 0.f};
        v8f acc[8];
#pragma unroll
        for (int nt = 0; nt < 8; ++nt) acc[nt] = zacc;

        // A fragment for k-step t (t = 0..63): t<32 from x, else from agg
        auto aload = [&](int t) -> v2f {
            return (t < 32) ? *(const v2f*)(px + 4 * t)
                            : *(const v2f*)(pa + 4 * (t - 32));
        };

        v2f bf[2][8];
#pragma unroll
        for (int nt = 0; nt < 8; ++nt)
            bf[0][nt] = *(const v2f*)(&ldsb[(nt * 16 + m) * LDSB_STRIDE_N + 2 * hi]);
        v2f a = aload(0);

#pragma unroll
        for (int t = 0; t < 64; ++t) {
            const int cur = t & 1, nxt = cur ^ 1;
            v2f an = a;
            if (t < 63) {
                const int kk = 4 * (t + 1) + 2 * hi;
#pragma unroll
                for (int nt = 0; nt < 8; ++nt)
                    bf[nxt][nt] = *(const v2f*)(&ldsb[(nt * 16 + m) * LDSB_STRIDE_N + kk]);
                an = aload(t + 1);
            }
            __builtin_amdgcn_sched_barrier(0);
#pragma unroll
            for (int nt = 0; nt < 8; ++nt)
                acc[nt] = __builtin_amdgcn_wmma_f32_16x16x4_f32(
                    false, a, false, bf[cur][nt], (short)0, acc[nt], false, false);
            a = an;
        }

#pragma unroll
        for (int r = 0; r < 8; ++r) {
            float* hrow = h + (long)(v0 + r + 8 * hi) * 128 + m;
#pragma unroll
            for (int nt = 0; nt < 8; ++nt) {
                float v = acc[nt][r] + bias[nt];
                hrow[nt * 16] = fmaxf(v, 0.0f);
            }
        }
    }
}

// ---------------------------------------------------------------------------
// Edge prediction head: one wave per prediction edge.
// emb = h[s] + h[d]; out[e] = relu(emb.ewp_w + ewp_b); out[E+e] = emb.ep_w + ep_b
// ---------------------------------------------------------------------------
__global__ __launch_bounds__(256) void edge_pred_kernel(
    const float* __restrict__ h,     // [N,128]
    const int*   __restrict__ ps,
    const int*   __restrict__ pd,
    const float* __restrict__ ewp_w, // [128]
    const float* __restrict__ ewp_b, // [1]
    const float* __restrict__ ep_w,  // [128]
    const float* __restrict__ ep_b,  // [1]
    float*       __restrict__ out,   // [2*E]
    int E)
{
    const int wave = threadIdx.x >> 5;
    const int lane = threadIdx.x & 31;
    const int e = blockIdx.x * 8 + wave;
    if (e >= E) return;

    const int s = ps[e], d = pd[e];
    const float4 a  = ((const float4*)(h + (long)s * 128))[lane];
    const float4 b  = ((const float4*)(h + (long)d * 128))[lane];
    const float4 w1 = ((const float4*)ewp_w)[lane];
    const float4 w2 = ((const float4*)ep_w)[lane];

    const float ex = a.x + b.x, ey = a.y + b.y, ez = a.z + b.z, ew = a.w + b.w;
    float dot1 = ex * w1.x + ey * w1.y + ez * w1.z + ew * w1.w;
    float dot2 = ex * w2.x + ey * w2.y + ez * w2.z + ew * w2.w;

#pragma unroll
    for (int off = 16; off >= 1; off >>= 1) {
        dot1 += __shfl_down(dot1, off, 32);
        dot2 += __shfl_down(dot2, off, 32);
    }
    if (lane == 0) {
        out[e]     = fmaxf(dot1 + ewp_b[0], 0.0f);
        out[E + e] = dot2 + ep_b[0];
    }
}

// ---------------------------------------------------------------------------
extern "C" void kernel_launch(void* const* d_in, const int* in_sizes, int n_in,
                              void* d_out, int out_size, void* d_ws, size_t ws_size,
                              hipStream_t stream) {
    const float* x     = (const float*)d_in[0];
    const int*   pe    = (const int*)  d_in[1];
    const int*   me    = (const int*)  d_in[2];
    const float* mwt   = (const float*)d_in[3];
    const float* c1pw  = (const float*)d_in[4];
    const float* c1pb  = (const float*)d_in[5];
    const float* c1fw  = (const float*)d_in[6];
    const float* c1fb  = (const float*)d_in[7];
    const float* c1cf  = (const float*)d_in[8];
    const float* c2pw  = (const float*)d_in[9];
    const float* c2pb  = (const float*)d_in[10];
    const float* c2fw  = (const float*)d_in[11];
    const float* c2fb  = (const float*)d_in[12];
    const float* c2cf  = (const float*)d_in[13];
    const float* ewpw  = (const float*)d_in[14];
    const float* ewpb  = (const float*)d_in[15];
    const float* epw   = (const float*)d_in[16];
    const float* epb   = (const float*)d_in[17];

    const int N  = in_sizes[0] / 128;
    const int EP = in_sizes[1] / 2;
    const int EM = in_sizes[2] / 2;
    const int* msrc = me;       const int* mdst = me + EM;
    const int* psrc = pe;       const int* pdst = pe + EP;

    float* agg = (float*)d_ws;
    float* h1  = agg + (size_t)N * 128;
    float* h2  = h1  + (size_t)N * 128;
    const size_t aggBytes = (size_t)N * 128 * sizeof(float);

    const int etiles = EM / 16;
    const int vtiles = N / 16;
    const dim3 blk(256);
    const int egrid = (etiles + 8 * TPW_E - 1) / (8 * TPW_E);
    const int vgrid = (vtiles + 8 * TPW_N - 1) / (8 * TPW_N);
    const int pgrid = (EP + 7) / 8;
    const size_t ldsE = 128 * LDSB_STRIDE_E * sizeof(float);
    const size_t ldsN = 128 * LDSB_STRIDE_N * sizeof(float);

    // Layer 1
    hipMemsetAsync(agg, 0, aggBytes, stream);
    edge_pool_kernel<<<egrid, blk, ldsE, stream>>>(x, msrc, mdst, mwt, c1pw, c1pb, c1cf, agg, etiles);
    node_update_kernel<<<vgrid, blk, ldsN, stream>>>(x, agg, c1fw, c1fb, h1, vtiles);
    // Layer 2
    hipMemsetAsync(agg, 0, aggBytes, stream);
    edge_pool_kernel<<<egrid, blk, ldsE, stream>>>(h1, msrc, mdst, mwt, c2pw, c2pb, c2cf, agg, etiles);
    node_update_kernel<<<vgrid, blk, ldsN, stream>>>(h1, agg, c2fw, c2fb, h2, vtiles);
    // Prediction head
    edge_pred_kernel<<<pgrid, blk, 0, stream>>>(h2, psrc, pdst, ewpw, ewpb, epw, epb, (float*)d_out, EP);
}